// GNNRefiner_18906446037567
// MI455X (gfx1250) — compile-verified
//
#include <hip/hip_runtime.h>

// ---------------------------------------------------------------------------
// GNNRefiner: 2-layer GCN (Fin=1, HIDDEN=64, Fout=1), B=8, N=20000, E=640000.
// Algebraic collapse: both aggregations are scalar scatters; only the
// relu(s*W1+b1)@W2 stage is 64-wide and is executed on the WMMA pipe
// (v_wmma_f32_16x16x4_f32, f32 end-to-end).
// ---------------------------------------------------------------------------

typedef float v2f __attribute__((ext_vector_type(2)));
typedef float v8f __attribute__((ext_vector_type(8)));

#define TPB 256

// deg[n] = 1.0 (self-loop weight)
__global__ void k_init_deg(float* __restrict__ deg, int n) {
    int i = blockIdx.x * blockDim.x + threadIdx.x;
    if (i < n) deg[i] = 1.0f;
}

// deg[col[e]] += ew[e]
__global__ void k_deg_scatter(const int* __restrict__ col, const float* __restrict__ ew,
                              float* __restrict__ deg, int e) {
    int i = blockIdx.x * blockDim.x + threadIdx.x;
    if (i < e) atomicAdd(&deg[col[i]], ew[i]);
}

// dis[n] = deg>0 ? rsqrt(deg) : 0   (in place)
__global__ void k_dis(float* __restrict__ deg, int n) {
    int i = blockIdx.x * blockDim.x + threadIdx.x;
    if (i < n) {
        float d = deg[i];
        deg[i] = (d > 0.0f) ? rsqrtf(d) : 0.0f;
    }
}

// xT[n*8+b] = x[b*N+n];  sT[n*8+b] = dis[n]^2 * x[b*N+n]  (self-loop term)
__global__ void k_xT_sinit(const float* __restrict__ x, const float* __restrict__ dis,
                           float* __restrict__ xT, float* __restrict__ sT, int n, int bn) {
    int i = blockIdx.x * blockDim.x + threadIdx.x;
    if (i < bn) {
        int b = i / n;
        int node = i - b * n;
        float xv = x[i];
        float dv = dis[node];
        xT[node * 8 + b] = xv;
        sT[node * 8 + b] = dv * dv * xv;
    }
}

// dstT[col*8+b] += dis[row]*ew*dis[col] * srcT[row*8+b]   (b = 0..7)
__global__ void k_edge_scatter(const int* __restrict__ row, const int* __restrict__ col,
                               const float* __restrict__ ew, const float* __restrict__ dis,
                               const float* __restrict__ srcT, float* __restrict__ dstT, int e) {
    int i = blockIdx.x * blockDim.x + threadIdx.x;
    if (i < e) {
        int r = row[i];
        int c = col[i];
        float w = dis[r] * ew[i] * dis[c];
        const float* sp = srcT + (size_t)r * 8;
        float* dp = dstT + (size_t)c * 8;
#pragma unroll
        for (int b = 0; b < 8; ++b) atomicAdd(&dp[b], w * sp[b]);
    }
}

// tT[i] = sum_f relu(sT[i]*W1[f] + b1[f]) * W2[f]
// One wave handles 16 rows; K=64 in 16 chained v_wmma_f32_16x16x4_f32 steps.
// B matrix = W2 replicated over all 16 columns -> every D column equals t[m].
__global__ void k_t_wmma(const float* __restrict__ sT, const float* __restrict__ W1,
                         const float* __restrict__ b1, const float* __restrict__ W2,
                         float* __restrict__ tT, int bn) {
    int wave = (int)((blockIdx.x * blockDim.x + threadIdx.x) >> 5);
    int lane = threadIdx.x & 31;
    int ntiles = bn >> 4;
    if (wave >= ntiles) return;  // uniform per wave: EXEC all-ones inside

    int base = wave * 16;
    int m = lane & 15;
    int koff = (lane < 16) ? 0 : 2;  // K half per lane group (A 16x4 layout)
    float sv = sT[base + m];

    v8f c = {0.f, 0.f, 0.f, 0.f, 0.f, 0.f, 0.f, 0.f};
#pragma unroll
    for (int k0 = 0; k0 < 64; k0 += 4) {
        int ka = k0 + koff;
        v2f a, b;
        a.x = fmaxf(fmaf(sv, W1[ka],     b1[ka]),     0.0f);
        a.y = fmaxf(fmaf(sv, W1[ka + 1], b1[ka + 1]), 0.0f);
        b.x = W2[ka];
        b.y = W2[ka + 1];
        c = __builtin_amdgcn_wmma_f32_16x16x4_f32(false, a, false, b,
                                                  (short)0, c, false, false);
    }
    // D layout: lane<16 -> M=vgpr (0..7), lane>=16 -> M=vgpr+8, N=lane&15.
    if (lane < 8) {
        tT[base + lane] = c[lane];                    // M = lane (at N = lane)
    } else if (lane >= 16 && lane < 24) {
        tT[base + 8 + (lane - 16)] = c[lane - 16];    // M = 8 + (lane-16)
    }
}

// deltaT[i] = dis[i/8]^2 * tT[i]   (self-loop term of 2nd aggregation)
__global__ void k_delta_init(const float* __restrict__ tT, const float* __restrict__ dis,
                             float* __restrict__ deltaT, int bn) {
    int i = blockIdx.x * blockDim.x + threadIdx.x;
    if (i < bn) {
        float d = dis[i >> 3];
        deltaT[i] = d * d * tT[i];
    }
}

// out[b*N+n] = x[b*N+n] + 0.5*(deltaT[n*8+b] + b2)
__global__ void k_final(const float* __restrict__ x, const float* __restrict__ deltaT,
                        const float* __restrict__ b2, float* __restrict__ out, int n, int bn) {
    int i = blockIdx.x * blockDim.x + threadIdx.x;
    if (i < bn) {
        int b = i / n;
        int node = i - b * n;
        out[i] = fmaf(0.5f, deltaT[node * 8 + b] + b2[0], x[i]);
    }
}

extern "C" void kernel_launch(void* const* d_in, const int* in_sizes, int n_in,
                              void* d_out, int out_size, void* d_ws, size_t ws_size,
                              hipStream_t stream) {
    const float* x   = (const float*)d_in[0];     // [B*N]
    const int*   ei  = (const int*)d_in[1];       // [2*E] (row | col)
    const float* ew  = (const float*)d_in[2];     // [E]
    const float* W1  = (const float*)d_in[3];     // [64]
    const float* b1  = (const float*)d_in[4];     // [64]
    const float* W2  = (const float*)d_in[5];     // [64]
    const float* b2  = (const float*)d_in[6];     // [1]
    float* out = (float*)d_out;

    const int BN = in_sizes[0];        // B*N = 160000
    const int B  = 8;
    const int N  = BN / B;             // 20000
    const int E  = in_sizes[2];        // 640000
    const int* row = ei;
    const int* col = ei + E;

    // Workspace layout (floats): deg/dis | xT | sT | tT | deltaT
    float* ws     = (float*)d_ws;
    float* dis    = ws;                        // N      (deg -> dis in place)
    float* xT     = ws + 20480;                // BN, [n][b]
    float* sT     = xT + BN;                   // BN
    float* tT     = sT + BN;                   // BN
    float* deltaT = tT + BN;                   // BN

    const int gN  = (N + TPB - 1) / TPB;
    const int gE  = (E + TPB - 1) / TPB;
    const int gBN = (BN + TPB - 1) / TPB;
    const int ntiles = BN / 16;                            // 10000
    const int gW  = (ntiles + (TPB / 32) - 1) / (TPB / 32); // waves per block = 8

    // 1) deg = 1 (self-loops), scatter edge weights, dis = rsqrt(deg)
    k_init_deg<<<gN, TPB, 0, stream>>>(dis, N);
    k_deg_scatter<<<gE, TPB, 0, stream>>>(col, ew, dis, E);
    k_dis<<<gN, TPB, 0, stream>>>(dis, N);

    // 2) transpose x, init s with self-loop term, scatter scalar messages
    k_xT_sinit<<<gBN, TPB, 0, stream>>>(x, dis, xT, sT, N, BN);
    k_edge_scatter<<<gE, TPB, 0, stream>>>(row, col, ew, dis, xT, sT, E);

    // 3) dense stage on the WMMA pipe: t = relu(s*W1 + b1) @ W2
    k_t_wmma<<<gW, TPB, 0, stream>>>(sT, W1, b1, W2, tT, BN);

    // 4) second aggregation (self-loop init + edge scatter), then output
    k_delta_init<<<gBN, TPB, 0, stream>>>(tT, dis, deltaT, BN);
    k_edge_scatter<<<gE, TPB, 0, stream>>>(row, col, ew, dis, tT, deltaT, E);
    k_final<<<gBN, TPB, 0, stream>>>(x, deltaT, b2, out, N, BN);
}